// RLayer_35081292874429
// MI455X (gfx1250) — compile-verified
//
#include <hip/hip_runtime.h>

// HIP's float4 is a struct; use a true vector type so __builtin_nontemporal_*
// and 128-bit codegen work cleanly.
typedef __attribute__((ext_vector_type(4))) float v4f;

#define DIM   8192
#define QPD   (DIM / 4)          // float4 quads per row = 2048
#define TPB   256                // 8 wave32 waves per block
#define NBLK  4096               // stride = NBLK*TPB = 1,048,576 = 512 * QPD

// Convert a generic pointer to a __shared__ object into the raw LDS byte
// offset (addrspace(3) pointers are 32-bit on amdgcn).
__device__ __forceinline__ unsigned lds_off_of(const void* p) {
  return (unsigned)(unsigned long long)(__attribute__((address_space(3))) const void*)p;
}

__global__ __launch_bounds__(TPB) void diag_scale_kernel(
    const v4f* __restrict__ x, const float* __restrict__ w,
    v4f* __restrict__ out, long long nquads) {
  __shared__ v4f wsh[QPD];  // 32 KiB: the whole weight vector, per block

  // Stage w into LDS with CDNA5 async global->LDS copies (ASYNCcnt path).
  // 2048 quads / 256 threads = 8 async b128 transfers per lane.
  const v4f* wq = (const v4f*)w;
  for (int i = threadIdx.x; i < QPD; i += TPB) {
    unsigned loff = lds_off_of(&wsh[i]);
    const v4f* gp = wq + i;
    asm volatile("global_load_async_to_lds_b128 %0, %1, off"
                 :: "v"(loff), "v"(gp)
                 : "memory");
  }
  asm volatile("s_wait_asynccnt 0" ::: "memory");
  __syncthreads();

  const long long stride = (long long)NBLK * TPB;  // multiple of QPD
  long long q = (long long)blockIdx.x * TPB + threadIdx.x;
  if (q >= nquads) return;

  // stride % QPD == 0, so this thread's column (and weight quad) is
  // loop-invariant: one ds_load_b128 total, hoisted out of the stream loop.
  const v4f wv = wsh[(int)(q & (QPD - 1))];

  // 16 trips/thread; unroll x4 so each wave keeps 4 independent NT b128
  // loads in flight (intra-wave MLP + load/store clauses), on top of the
  // ~32K-wave TLP that actually saturates the 23.3 TB/s HBM.
#pragma unroll 4
  for (; q < nquads; q += stride) {
    // x is read exactly once and out written exactly once: use non-temporal
    // hints so the 512 MiB stream doesn't evict useful lines from L2.
    v4f xv = __builtin_nontemporal_load(&x[q]);
    v4f r  = xv * wv;
    __builtin_nontemporal_store(r, &out[q]);
  }
}

extern "C" void kernel_launch(void* const* d_in, const int* in_sizes, int n_in,
                              void* d_out, int out_size, void* d_ws, size_t ws_size,
                              hipStream_t stream) {
  const v4f*  x = (const v4f*)d_in[0];         // (N, DIM) fp32
  const float* w = (const float*)d_in[1];      // (DIM,)  fp32
  v4f* out = (v4f*)d_out;                      // (N, DIM) fp32

  long long nquads = (long long)in_sizes[0] / 4;  // 16,777,216
  diag_scale_kernel<<<NBLK, TPB, 0, stream>>>(x, w, out, nquads);
}